// gcnBlock_24730421690785
// MI455X (gfx1250) — compile-verified
//
#include <hip/hip_runtime.h>
#include <hip/hip_bf16.h>

typedef float v2f __attribute__((ext_vector_type(2)));
typedef float v8f __attribute__((ext_vector_type(8)));

#define DF 128
#define LN_EPS 1e-5f

// ---------------------------------------------------------------------------
// 1) zero the output accumulator and degree counters
// ---------------------------------------------------------------------------
__global__ void gcn_zero_kernel(float* __restrict__ out, unsigned* __restrict__ cnt, int N) {
    long long i = (long long)blockIdx.x * blockDim.x + threadIdx.x;
    long long total = (long long)N * DF;
    if (i < total) out[i] = 0.0f;
    if (i < N) cnt[i] = 0u;
}

// ---------------------------------------------------------------------------
// 2) degree count on dst (self-loops folded in later as +1)
// ---------------------------------------------------------------------------
__global__ void gcn_deg_kernel(const int* __restrict__ ei, unsigned* __restrict__ cnt, int E) {
    int e = blockIdx.x * blockDim.x + threadIdx.x;
    if (e < E) {
        int dst = ei[E + e];
        atomicAdd(&cnt[dst], 1u);
    }
}

// ---------------------------------------------------------------------------
// 3) dinv = rsqrt(deg) in place (deg = cnt + 1 self-loop, always > 0)
// ---------------------------------------------------------------------------
__global__ void gcn_dinv_kernel(unsigned* __restrict__ cnt, int N) {
    int i = blockIdx.x * blockDim.x + threadIdx.x;
    if (i < N) {
        float deg = (float)cnt[i] + 1.0f;
        ((float*)cnt)[i] = rsqrtf(deg);
    }
}

// ---------------------------------------------------------------------------
// 4) h = x @ W via v_wmma_f32_16x16x4_f32. One wave per 16x16 output tile.
//    A(16x4): lanes 0-15 hold K={k,k+1} for M=lane; lanes 16-31 hold K={k+2,k+3}.
//    B(4x16): lanes 0-15 hold rows K={k,k+1} at N=lane; lanes 16-31 rows {k+2,k+3}.
//    C/D(16x16): VGPR r holds M=r (lanes 0-15) and M=8+r (lanes 16-31).
// ---------------------------------------------------------------------------
__global__ void gcn_gemm_wmma_kernel(const float* __restrict__ x,
                                     const float* __restrict__ W,
                                     float* __restrict__ h, int N) {
    int wave = (int)((blockIdx.x * blockDim.x + threadIdx.x) >> 5);
    int lane = threadIdx.x & 31;
    int rowTiles = (N + 15) >> 4;
    if (wave >= rowTiles * (DF / 16)) return;   // wave-uniform: EXEC stays all-ones

    int rt = wave >> 3;       // row tile (16 rows of x)
    int ct = wave & 7;        // col tile (16 cols of W)
    int half = lane >> 4;
    int lidx = lane & 15;

    int row = rt * 16 + lidx;
    if (row >= N) row = N - 1;                  // clamp reads only
    const float* xr = x + (size_t)row * DF;
    int ncol = ct * 16 + lidx;

    v8f c = {};
#pragma unroll
    for (int kk = 0; kk < DF; kk += 4) {
        int k = kk + 2 * half;
        v2f a, bm;
        a.x  = xr[k];
        a.y  = xr[k + 1];
        bm.x = W[(size_t)k * DF + ncol];
        bm.y = W[(size_t)(k + 1) * DF + ncol];
        c = __builtin_amdgcn_wmma_f32_16x16x4_f32(
                /*neg_a=*/false, a, /*neg_b=*/false, bm,
                /*c_mod=*/(short)0, c, /*reuse_a=*/false, /*reuse_b=*/false);
    }

#pragma unroll
    for (int r = 0; r < 8; ++r) {
        int orow = rt * 16 + half * 8 + r;
        if (orow < N) h[(size_t)orow * DF + ncol] = c[r];
    }
}

// ---------------------------------------------------------------------------
// 5) edge scatter: one wave per edge, lane handles 4 contiguous features.
//    out[dst] += h[src] * dinv[src]*dinv[dst]   (f32 atomics, L2-resident)
// ---------------------------------------------------------------------------
__global__ void gcn_scatter_kernel(const int* __restrict__ ei,
                                   const float* __restrict__ h,
                                   const float* __restrict__ dinv,
                                   float* __restrict__ out, int E) {
    long long tid = (long long)blockIdx.x * blockDim.x + threadIdx.x;
    long long e = tid >> 5;
    int lane = (int)(tid & 31);
    if (e >= E) return;

    int src = ei[e];
    int dst = ei[E + e];
    float norm = dinv[src] * dinv[dst];

    float4 v = ((const float4*)(h + (size_t)src * DF))[lane];
    float* op = out + (size_t)dst * DF + lane * 4;
    atomicAdd(op + 0, v.x * norm);
    atomicAdd(op + 1, v.y * norm);
    atomicAdd(op + 2, v.z * norm);
    atomicAdd(op + 3, v.w * norm);
}

// ---------------------------------------------------------------------------
// 6) fused self-loop + bias + LayerNorm + ReLU, one wave32 per row
// ---------------------------------------------------------------------------
__global__ void gcn_ln_kernel(const float* __restrict__ h,
                              const float* __restrict__ dinv,
                              const float* __restrict__ bias,
                              const float* __restrict__ gamma,
                              const float* __restrict__ beta,
                              float* __restrict__ out, int N) {
    int row  = (int)((blockIdx.x * blockDim.x + threadIdx.x) >> 5);
    int lane = threadIdx.x & 31;
    if (row >= N) return;      // wave-uniform

    float di = dinv[row];
    float selfn = di * di;

    float4 acc = ((const float4*)(out + (size_t)row * DF))[lane];
    float4 hv  = ((const float4*)(h   + (size_t)row * DF))[lane];
    float4 bv  = ((const float4*)bias)[lane];

    float v0 = acc.x + hv.x * selfn + bv.x;
    float v1 = acc.y + hv.y * selfn + bv.y;
    float v2 = acc.z + hv.z * selfn + bv.z;
    float v3 = acc.w + hv.w * selfn + bv.w;

    float s  = v0 + v1 + v2 + v3;
    float sq = v0 * v0 + v1 * v1 + v2 * v2 + v3 * v3;
#pragma unroll
    for (int off = 16; off > 0; off >>= 1) {
        s  += __shfl_xor(s,  off, 32);
        sq += __shfl_xor(sq, off, 32);
    }
    float mu   = s * (1.0f / DF);
    float var  = sq * (1.0f / DF) - mu * mu;
    float rstd = rsqrtf(var + LN_EPS);

    float4 gv = ((const float4*)gamma)[lane];
    float4 tv = ((const float4*)beta)[lane];
    float4 o;
    o.x = fmaxf(0.0f, (v0 - mu) * rstd * gv.x + tv.x);
    o.y = fmaxf(0.0f, (v1 - mu) * rstd * gv.y + tv.y);
    o.z = fmaxf(0.0f, (v2 - mu) * rstd * gv.z + tv.z);
    o.w = fmaxf(0.0f, (v3 - mu) * rstd * gv.w + tv.w);

    ((float4*)(out + (size_t)row * DF))[lane] = o;
}

// ---------------------------------------------------------------------------
extern "C" void kernel_launch(void* const* d_in, const int* in_sizes, int n_in,
                              void* d_out, int out_size, void* d_ws, size_t ws_size,
                              hipStream_t stream) {
    const float* x     = (const float*)d_in[0];
    const int*   ei    = (const int*)d_in[1];
    const float* W     = (const float*)d_in[2];
    const float* bias  = (const float*)d_in[3];
    const float* gamma = (const float*)d_in[4];
    const float* beta  = (const float*)d_in[5];
    float* out = (float*)d_out;

    int N = in_sizes[0] / DF;
    int E = in_sizes[1] / 2;

    float*    h   = (float*)d_ws;                                    // N*128 f32
    unsigned* cnt = (unsigned*)((char*)d_ws + (size_t)N * DF * 4);   // N u32 -> f32 dinv

    long long total = (long long)N * DF;
    gcn_zero_kernel<<<(int)((total + 255) / 256), 256, 0, stream>>>(out, cnt, N);
    gcn_deg_kernel<<<(E + 255) / 256, 256, 0, stream>>>(ei, cnt, E);
    gcn_dinv_kernel<<<(N + 255) / 256, 256, 0, stream>>>(cnt, N);

    int rowTiles = (N + 15) / 16;
    int tiles = rowTiles * (DF / 16);
    gcn_gemm_wmma_kernel<<<(tiles + 7) / 8, 256, 0, stream>>>(x, W, h, N);

    long long sthreads = (long long)E * 32;
    gcn_scatter_kernel<<<(int)((sthreads + 255) / 256), 256, 0, stream>>>(
        ei, h, (const float*)cnt, out, E);

    gcn_ln_kernel<<<(N + 7) / 8, 256, 0, stream>>>(
        h, (const float*)cnt, bias, gamma, beta, out, N);
}